// ApertureChamberSSM_64828236366560
// MI455X (gfx1250) — compile-verified
//
#include <hip/hip_runtime.h>
#include <math.h>

// ---------------------------------------------------------------------------
// ApertureChamberSSM: gated tanh-drive linear scan along S + half-dim rotation
// x: (B=4, S=8192, D=1024) fp32.  Memory-bound (~268MB min traffic).
// Two-kernel chunked parallel scan (constant decay => carry is just the
// initial state of the recurrence; chunk-combine factor is decay^L).
// ---------------------------------------------------------------------------

#define AS1 __attribute__((address_space(1)))
#define AS3 __attribute__((address_space(3)))

static constexpr int B    = 4;
static constexpr int S    = 8192;
static constexpr int D    = 1024;
static constexpr int HALF = D / 2;        // 512
static constexpr int NC   = 64;           // chunks along S
static constexpr int CL   = S / NC;       // 128 timesteps per chunk
static constexpr int STEPS  = 8;          // timesteps per async stage (kernel 2)
static constexpr int STAGES = CL / STEPS; // 16

__device__ __forceinline__ float sigf(float v) { return 1.0f / (1.0f + expf(-v)); }

__device__ __forceinline__ float fast_tanh(float v) {
#if __has_builtin(__builtin_amdgcn_tanhf)
  return __builtin_amdgcn_tanhf(v);      // gfx1250 v_tanh_f32
#elif __has_builtin(__builtin_amdgcn_tanh_f32)
  return __builtin_amdgcn_tanh_f32(v);
#else
  return tanhf(v);
#endif
}

__device__ __forceinline__ void async_copy_b32(const float* gsrc, float* lds_dst) {
  __builtin_amdgcn_global_load_async_to_lds_b32(
      (AS1 int*)(unsigned long long)gsrc,
      (AS3 int*)lds_dst, 0, 0);
}

template <int N>
__device__ __forceinline__ void wait_asynccnt() {
#if __has_builtin(__builtin_amdgcn_s_wait_asynccnt)
  __builtin_amdgcn_s_wait_asynccnt(N);
#else
  asm volatile("s_wait_asynccnt %0" :: "n"(N) : "memory");
#endif
}

// ---------------------------------------------------------------------------
// Kernel 1: per-(b,d,chunk) local scan end value E_c (zero-initialized state).
// grid = B*NC*D/256 blocks of 256. Lanes stride d => coalesced.
// ---------------------------------------------------------------------------
__global__ __launch_bounds__(256)
void k_partials(const float* __restrict__ x,
                const float* __restrict__ p_alpha,
                const float* __restrict__ p_iv,
                float* __restrict__ partials)
{
  const float alpha = sigf(p_alpha[0]);
  const float iv    = sigf(p_iv[0]);
  const float decay = expf(-alpha);
  const float w     = 1.0f - decay;

  const int gid = blockIdx.x * 256 + threadIdx.x;     // [0, B*NC*D)
  const int d   = gid % D;
  const int bc  = gid / D;                            // b*NC + c
  const int c   = bc % NC;
  const int b   = bc / NC;

  const float* xp = x + ((size_t)b * S + (size_t)c * CL) * D + d;

  float p = 0.0f;
#pragma unroll 8
  for (int s = 0; s < CL; ++s) {
    const float drive = fast_tanh(iv * xp[(size_t)s * D]);
    p = fmaf(decay, p, w * drive);
  }
  partials[(size_t)bc * D + d] = p;                   // [(b*NC + c)*D + d]
}

// ---------------------------------------------------------------------------
// Kernel 2: fold carries, rerun chunk with carry-initialized state, apply gate
// mix + rotation (pairs d and d+512 live in one thread) + output valve.
// x tile is staged through LDS with gfx1250 async global->LDS loads
// (double-buffered, throttled with s_wait_asynccnt) when available.
// grid = B*NC*(HALF/256) blocks of 256.
// ---------------------------------------------------------------------------
#define HAS_ASYNC_LDS (__has_builtin(__builtin_amdgcn_global_load_async_to_lds_b32))

__global__ __launch_bounds__(256)
void k_final(const float* __restrict__ x,
             const float* __restrict__ p_beta,
             const float* __restrict__ p_iv,
             const float* __restrict__ p_ov,
             const float* __restrict__ p_alpha,
             const float* __restrict__ p_mg,
             const float* __restrict__ partials,
             float* __restrict__ out)
{
  const float beta_s = sigf(p_beta[0]);
  const float iv     = sigf(p_iv[0]);
  const float ov     = sigf(p_ov[0]);
  const float alpha  = sigf(p_alpha[0]);
  const float mg     = sigf(p_mg[0]);
  const float decay  = expf(-alpha);
  const float w      = 1.0f - decay;
  const float angle  = 3.14159265358979323846f * beta_s;
  const float ca     = cosf(angle);
  const float sa     = sinf(angle);

  const int tid  = threadIdx.x;
  const int blk  = blockIdx.x;
  const int dblk = blk % (HALF / 256);                 // 0..1
  const int c    = (blk / (HALF / 256)) % NC;
  const int b    = blk / ((HALF / 256) * NC);
  const int d0   = dblk * 256 + tid;                   // [0, 512)
  const int d1   = d0 + HALF;

  // Carry-in C_c = sum_j decay^(CL*(c-1-j)) * E_j  (fold, ascending j)
  const float dL = expf(-alpha * (float)CL);
  float C0 = 0.0f, C1 = 0.0f;
  const float* pp = partials + (size_t)b * NC * D;
  for (int j = 0; j < c; ++j) {
    C0 = fmaf(C0, dL, pp[(size_t)j * D + d0]);
    C1 = fmaf(C1, dL, pp[(size_t)j * D + d1]);
  }
  float p0 = C0, p1 = C1;   // recurrence state initialized with carry

  const size_t base = ((size_t)b * S + (size_t)c * CL) * D;
  const float* xr = x + base + d0;
  const float* xi = x + base + d1;
  float* orp = out + base + d0;
  float* oip = out + base + d1;

#if HAS_ASYNC_LDS
  __shared__ float smem[2][STEPS][2][256];             // 32 KB, bank-conflict free

  // Prologue: stage 0
#pragma unroll
  for (int t = 0; t < STEPS; ++t) {
    async_copy_b32(xr + (size_t)t * D, &smem[0][t][0][tid]);
    async_copy_b32(xi + (size_t)t * D, &smem[0][t][1][tid]);
  }

  for (int st = 0; st < STAGES; ++st) {
    const int cur = st & 1;
    const int nxt = cur ^ 1;
    if (st + 1 < STAGES) {
      const float* xr2 = xr + (size_t)(st + 1) * STEPS * D;
      const float* xi2 = xi + (size_t)(st + 1) * STEPS * D;
#pragma unroll
      for (int t = 0; t < STEPS; ++t) {
        async_copy_b32(xr2 + (size_t)t * D, &smem[nxt][t][0][tid]);
        async_copy_b32(xi2 + (size_t)t * D, &smem[nxt][t][1][tid]);
      }
      // allow next stage (2*STEPS instrs) in flight; current stage complete
      wait_asynccnt<2 * STEPS>();
    } else {
      wait_asynccnt<0>();
    }

    float* op_r = orp + (size_t)st * STEPS * D;
    float* op_i = oip + (size_t)st * STEPS * D;
#pragma unroll
    for (int t = 0; t < STEPS; ++t) {
      const float vr  = smem[cur][t][0][tid];
      const float vi  = smem[cur][t][1][tid];
      const float inr = iv * vr;
      const float ini = iv * vi;
      p0 = fmaf(decay, p0, w * fast_tanh(inr));
      p1 = fmaf(decay, p1, w * fast_tanh(ini));
      const float mr = fmaf(mg, p0 - inr, inr);        // mg*psi + (1-mg)*x_in
      const float mi = fmaf(mg, p1 - ini, ini);
      op_r[(size_t)t * D] = (mr * ca - mi * sa) * ov;
      op_i[(size_t)t * D] = (mr * sa + mi * ca) * ov;
    }
  }
#else
  // Fallback: direct global loads, unrolled so loads pipeline ahead of the
  // serial FMA chain.
#pragma unroll 4
  for (int s = 0; s < CL; ++s) {
    const float vr  = xr[(size_t)s * D];
    const float vi  = xi[(size_t)s * D];
    const float inr = iv * vr;
    const float ini = iv * vi;
    p0 = fmaf(decay, p0, w * fast_tanh(inr));
    p1 = fmaf(decay, p1, w * fast_tanh(ini));
    const float mr = fmaf(mg, p0 - inr, inr);
    const float mi = fmaf(mg, p1 - ini, ini);
    orp[(size_t)s * D] = (mr * ca - mi * sa) * ov;
    oip[(size_t)s * D] = (mr * sa + mi * ca) * ov;
  }
#endif
}

// ---------------------------------------------------------------------------
// Host launcher
// ---------------------------------------------------------------------------
extern "C" void kernel_launch(void* const* d_in, const int* in_sizes, int n_in,
                              void* d_out, int out_size, void* d_ws, size_t ws_size,
                              hipStream_t stream)
{
  const float* x       = (const float*)d_in[0];
  const float* beta    = (const float*)d_in[1];
  const float* in_v    = (const float*)d_in[2];
  const float* out_v   = (const float*)d_in[3];
  const float* alpha_r = (const float*)d_in[4];
  const float* mem_g   = (const float*)d_in[5];
  float* out           = (float*)d_out;
  float* partials      = (float*)d_ws;   // B*NC*D floats = 1 MB scratch

  (void)in_sizes; (void)n_in; (void)out_size; (void)ws_size;

  k_partials<<<B * NC * D / 256, 256, 0, stream>>>(x, alpha_r, in_v, partials);
  k_final<<<B * NC * (HALF / 256), 256, 0, stream>>>(
      x, beta, in_v, out_v, alpha_r, mem_g, partials, out);
}